// TAdaConv2d_4947802325756
// MI455X (gfx1250) — compile-verified
//
#include <hip/hip_runtime.h>

typedef __attribute__((ext_vector_type(2))) float v2f;
typedef __attribute__((ext_vector_type(8))) float v8f;

#if defined(__has_builtin)
#  if __has_builtin(__builtin_amdgcn_global_load_async_to_lds_b32)
#    define HAVE_ASYNC_LDS 1
#  endif
#endif

#define NB   8
#define CI   64
#define NT   16
#define HH   64
#define WW   64
#define CO   64
#define HW   (HH*WW)
#define CHUNK 16            // channels per K-chunk
#define KC   (CHUNK*9)      // 144 K per chunk (tap-major: k = tap*16 + ci_local)
#define KP   145            // padded K stride for Ws (odd -> conflict-free)
#define CIP  17             // padded channel stride for Xs (conflict-free)
#define TS   8              // 8x8 spatial tile

__global__ __launch_bounds__(128) void tada_conv2d_wmma_f32(
    const float* __restrict__ x,      // (b, ci, t, h, w)
    const float* __restrict__ alpha,  // (b, ci, t, 1, 1)
    const float* __restrict__ weight, // (co, ci, 3, 3)
    const float* __restrict__ bias,   // (co)
    float* __restrict__ out)          // (b, co, t, h, w)
{
    __shared__ float Ws[CO * KP];                 // scaled weights, [co][k]
    __shared__ float Xs[(TS+2)*(TS+2)*CIP];       // input tile w/ halo, [y][x][ci]

    const int tid  = threadIdx.x;
    const int wave = tid >> 5;
    const int lane = tid & 31;
    const int lm   = lane & 15;    // A-row (M) / B-col (N) index
    const int hi   = lane >> 4;    // K half: lanes 16-31 hold K+2,K+3

    const int bt   = blockIdx.x >> 6;   // 64 tiles per (b,t) image
    const int tile = blockIdx.x & 63;
    const int b    = bt / NT;
    const int t    = bt - b * NT;
    const int y0   = (tile >> 3) * TS;
    const int x0   = (tile & 7) * TS;

    const int cobase = wave * 16;

    // pixel bases for the 4 N-tiles: p = j*16+lm -> py = j*2 + (lm>>3), px = lm&7
    const int pb0 = ((0*2 + (lm >> 3)) * 10) + (lm & 7);
    const int pb1 = ((1*2 + (lm >> 3)) * 10) + (lm & 7);
    const int pb2 = ((2*2 + (lm >> 3)) * 10) + (lm & 7);
    const int pb3 = ((3*2 + (lm >> 3)) * 10) + (lm & 7);

    v8f acc[4] = {v8f{}, v8f{}, v8f{}, v8f{}};

    for (int ch = 0; ch < CI / CHUNK; ++ch) {
        const int cc = ch * CHUNK;
        __syncthreads();   // previous chunk's compute done before overwrite

        // ---- stage alpha-scaled weights: Ws[co][tap*16+cil] (needs the multiply,
        //      so it keeps the register path) ----
        for (int idx = tid; idx < CO * KC; idx += 128) {
            const int co  = idx / KC;
            const int k   = idx - co * KC;
            const int tap = k >> 4;
            const int cil = k & 15;
            const int ci  = cc + cil;
            const float wv = weight[(co * CI + ci) * 9 + tap];
            const float av = alpha[(b * CI + ci) * NT + t];
            Ws[co * KP + k] = wv * av;
        }

        // ---- stage zero-padded input tile: Xs[ly][lx][cil] (1600 elems) ----
#ifdef HAVE_ASYNC_LDS
        // CDNA5 async memory->LDS path (ASYNCcnt), no VGPR round-trip.
        for (int idx = tid; idx < CHUNK * 100; idx += 128) {
            const int cil = idx / 100;
            const int pos = idx - cil * 100;
            const int ly  = pos / 10;
            const int lx  = pos - ly * 10;
            const int gy  = y0 - 1 + ly;
            const int gx  = x0 - 1 + lx;
            const int ldso = (ly * 10 + lx) * CIP + cil;
            if (gy >= 0 && gy < HH && gx >= 0 && gx < WW) {
                const float* gp =
                    x + ((size_t)(b * CI + cc + cil) * NT + t) * HW + gy * WW + gx;
                __builtin_amdgcn_global_load_async_to_lds_b32(
                    (__attribute__((address_space(1))) int*)gp,
                    (__attribute__((address_space(3))) int*)&Xs[ldso],
                    0, 0);
            } else {
                Xs[ldso] = 0.0f;    // halo zero-padding for EXEC-masked lanes
            }
        }
#  if __has_builtin(__builtin_amdgcn_s_wait_asynccnt)
        __builtin_amdgcn_s_wait_asynccnt(0);
#  else
        asm volatile("s_wait_asynccnt 0" ::: "memory");
#  endif
#else
        for (int idx = tid; idx < CHUNK * 100; idx += 128) {
            const int cil = idx / 100;
            const int pos = idx - cil * 100;
            const int ly  = pos / 10;
            const int lx  = pos - ly * 10;
            const int gy  = y0 - 1 + ly;
            const int gx  = x0 - 1 + lx;
            float v = 0.0f;
            if (gy >= 0 && gy < HH && gx >= 0 && gx < WW)
                v = x[((size_t)(b * CI + cc + cil) * NT + t) * HW + gy * WW + gx];
            Xs[(ly * 10 + lx) * CIP + cil] = v;
        }
#endif
        __syncthreads();

        // ---- prefetch next chunk's tile rows (overlaps with compute below);
        //      one prefetch per (ci,row): a cacheline covers the 10-float row ----
        if (ch + 1 < CI / CHUNK) {
            const int cc2 = cc + CHUNK;
            for (int idx = tid; idx < CHUNK * 10; idx += 128) {
                const int cil = idx / 10;
                const int ly  = idx - cil * 10;
                const int gy  = y0 - 1 + ly;
                if (gy >= 0 && gy < HH) {
                    const int gx = (x0 > 0) ? (x0 - 1) : 0;
                    __builtin_prefetch(
                        x + ((size_t)(b * CI + cc2 + cil) * NT + t) * HW + gy * WW + gx,
                        0, 1);
                }
            }
        }

        // ---- compute: 36 K-steps of 4, tap constant within each step ----
        const float* Arow = &Ws[(cobase + lm) * KP + 2 * hi];
        const int cioff = 2 * hi;
#pragma unroll
        for (int step = 0; step < KC / 4; ++step) {
            const int kb  = step * 4;
            const int tap = kb >> 4;         // compile-time after unroll
            const int dy  = tap / 3;
            const int dx  = tap - dy * 3;
            const int cib = kb & 15;

            v2f A;
            A.x = Arow[kb];
            A.y = Arow[kb + 1];

            const int xoff = (dy * 10 + dx) * CIP + cib + cioff;
            {
                v2f Bv; const int a = pb0 * CIP + xoff;
                Bv.x = Xs[a]; Bv.y = Xs[a + 1];
                acc[0] = __builtin_amdgcn_wmma_f32_16x16x4_f32(
                    false, A, false, Bv, (short)0, acc[0], false, false);
            }
            {
                v2f Bv; const int a = pb1 * CIP + xoff;
                Bv.x = Xs[a]; Bv.y = Xs[a + 1];
                acc[1] = __builtin_amdgcn_wmma_f32_16x16x4_f32(
                    false, A, false, Bv, (short)0, acc[1], false, false);
            }
            {
                v2f Bv; const int a = pb2 * CIP + xoff;
                Bv.x = Xs[a]; Bv.y = Xs[a + 1];
                acc[2] = __builtin_amdgcn_wmma_f32_16x16x4_f32(
                    false, A, false, Bv, (short)0, acc[2], false, false);
            }
            {
                v2f Bv; const int a = pb3 * CIP + xoff;
                Bv.x = Xs[a]; Bv.y = Xs[a + 1];
                acc[3] = __builtin_amdgcn_wmma_f32_16x16x4_f32(
                    false, A, false, Bv, (short)0, acc[3], false, false);
            }
        }
    }

    // ---- epilogue: C/D layout -> (b, co, t, y, x), add bias ----
#pragma unroll
    for (int j = 0; j < 4; ++j) {
        const int p  = j * 16 + lm;
        const int py = p >> 3;
        const int px = p & 7;
        const size_t pix = (size_t)(y0 + py) * WW + (x0 + px);
#pragma unroll
        for (int e = 0; e < 8; ++e) {
            const int co = cobase + e + 8 * hi;   // VGPR e: M=e (lanes 0-15) / e+8 (16-31)
            out[((size_t)(b * CO + co) * NT + t) * HW + pix] = acc[j][e] + bias[co];
        }
    }
}

extern "C" void kernel_launch(void* const* d_in, const int* in_sizes, int n_in,
                              void* d_out, int out_size, void* d_ws, size_t ws_size,
                              hipStream_t stream) {
    const float* x      = (const float*)d_in[0];
    const float* alpha  = (const float*)d_in[1];
    const float* weight = (const float*)d_in[2];
    const float* bias   = (const float*)d_in[3];
    float* out          = (float*)d_out;

    dim3 grid(NB * NT * 64);   // 8192 workgroups
    dim3 block(128);           // 4 wave32
    hipLaunchKernelGGL(tada_conv2d_wmma_f32, grid, block, 0, stream,
                       x, alpha, weight, bias, out);
}